// SNN_42245298324152
// MI455X (gfx1250) — compile-verified
//
#include <hip/hip_runtime.h>
#include <stdint.h>

namespace {
constexpr int   T_STEPS = 1000;
constexpr int   NI      = 700;
constexpr int   NH      = 4096;
constexpr int   NO      = 20;
constexpr float ALPHA   = 0.8187307530779818f;  // exp(-0.001/0.005)
constexpr float BETA    = 0.9048374180359595f;  // exp(-0.001/0.01)
constexpr float TH      = 1.0f;
}

typedef __attribute__((ext_vector_type(2))) float v2f;
typedef __attribute__((ext_vector_type(8))) float v8f;

// ---------------------------------------------------------------------------
// Phase 0: zero the LIF state block (mem1, syn1, out1_a, out1_b, mem2, syn2)
// ---------------------------------------------------------------------------
__global__ __launch_bounds__(256) void snn_init(float* __restrict__ state, int n) {
  int i = blockIdx.x * 256 + threadIdx.x;
  if (i < n) state[i] = 0.0f;
}

// ---------------------------------------------------------------------------
// Phase 1: Cin[m, t] = sum_k W_in[m, k] * spikes[t, k]
// Dense fp32 GEMM [NH x NI] x [NI x T] via V_WMMA_F32_16X16X4_F32.
// One wave per 16x16 output tile; K marched in steps of 4.
//   A frag (16x4):  lane holds M = lane&15, K = 2*(lane>>4) + {0,1}  (v2f load)
//   B frag (4x16):  lane holds N = lane&15, K = 2*(lane>>4) + {0,1}  (v2f load)
//   C/D  (16x16):   vgpr r holds (M = r + 8*(lane>>4), N = lane&15)
// ---------------------------------------------------------------------------
__global__ __launch_bounds__(32) void snn_in_gemm(const float* __restrict__ Win,
                                                  const float* __restrict__ spikes,
                                                  float* __restrict__ Cin) {
  const int lane  = threadIdx.x;
  const int m0    = blockIdx.x * 16;
  const int t0    = blockIdx.y * 16;
  const int sub   = lane & 15;
  const int khalf = (lane >> 4) * 2;

  const int  mrow   = m0 + sub;
  const int  tcol   = t0 + sub;
  const bool tvalid = (tcol < T_STEPS);
  const int  tcl    = tvalid ? tcol : (T_STEPS - 1);
  const float bmask = tvalid ? 1.0f : 0.0f;   // zero out clamped tail columns

  const float* arow = Win    + (size_t)mrow * NI + khalf;
  const float* brow = spikes + (size_t)tcl  * NI + khalf;

  v8f c = {0.f, 0.f, 0.f, 0.f, 0.f, 0.f, 0.f, 0.f};
  for (int k = 0; k < NI; k += 4) {                 // 175 WMMA per tile
    v2f a = *(const v2f*)(arow + k);
    v2f b = *(const v2f*)(brow + k);
    b = b * bmask;                                  // uniform mask, no EXEC change
    c = __builtin_amdgcn_wmma_f32_16x16x4_f32(false, a, false, b,
                                              (short)0, c, false, false);
  }

  if (tvalid) {
    const int mb = m0 + ((lane >> 4) << 3);
#pragma unroll
    for (int r = 0; r < 8; ++r)
      Cin[(size_t)(mb + r) * T_STEPS + tcol] = c[r];
  }
}

// ---------------------------------------------------------------------------
// Phase 2a: hidden layer, one time step.
// y = W_fb @ out1_prev  (L2-resident 64MB weight matrix), then LIF update.
// Spike vector staged memory->LDS with GLOBAL_LOAD_ASYNC_TO_LDS_B128
// (ASYNCcnt path, no VGPR round-trip), one wave per row (8 rows / block),
// float4-coalesced row streaming, wave32 xor-shuffle reduction.
// ---------------------------------------------------------------------------
__global__ __launch_bounds__(256) void snn_hidden(const float* __restrict__ Wfb,
                                                  const float* __restrict__ Cin,
                                                  float* __restrict__ mem1,
                                                  float* __restrict__ syn1,
                                                  const float* __restrict__ sprev,
                                                  float* __restrict__ snext,
                                                  int t) {
  __shared__ float s_sp[NH];
  const int tid  = threadIdx.x;
  const int wave = tid >> 5;
  const int lane = tid & 31;
  const int row  = blockIdx.x * 8 + wave;

  // Warm the L2/L0 line holding this row's precomputed input current early.
  __builtin_prefetch(Cin + (size_t)row * T_STEPS + t, 0, 0);

  // Async stage 16KB of spikes into LDS: 256 lanes x 16B x 4 passes.
  // Flat LDS address truncates to the LDS byte offset in the low 32 bits.
  {
    const uint32_t lds_base = (uint32_t)(uintptr_t)s_sp;
    const uint64_t gbase    = (uint64_t)(uintptr_t)sprev;
#pragma unroll
    for (int it = 0; it < (NH * 4) / (256 * 16); ++it) {
      const uint32_t byteoff  = (uint32_t)(tid * 16 + it * 4096);
      const uint32_t lds_addr = lds_base + byteoff;
      const uint64_t gaddr    = gbase + byteoff;
      asm volatile("global_load_async_to_lds_b128 %0, %1, off"
                   :: "v"(lds_addr), "v"(gaddr)
                   : "memory");
    }
    asm volatile("s_wait_asynccnt 0x0" ::: "memory");
  }
  __syncthreads();

  const float4* w4 = (const float4*)(Wfb + (size_t)row * NH);
  const float4* s4 = (const float4*)s_sp;

  float acc = 0.0f;
#pragma unroll 4
  for (int i = lane; i < NH / 4; i += 32) {
    const float4 w = w4[i];
    const float4 s = s4[i];
    acc = fmaf(w.x, s.x, acc);
    acc = fmaf(w.y, s.y, acc);
    acc = fmaf(w.z, s.z, acc);
    acc = fmaf(w.w, s.w, acc);
  }
#pragma unroll
  for (int off = 16; off > 0; off >>= 1) acc += __shfl_xor(acc, off, 32);

  if (lane == 0) {
    const float cur    = Cin[(size_t)row * T_STEPS + t] + acc;
    const float sp     = s_sp[row];                 // rst = out1_prev
    const float m      = mem1[row] * (1.0f - sp);
    const float newout = ((m - TH) > 0.0f) ? 1.0f : 0.0f;
    const float s1     = syn1[row];
    mem1[row]  = fmaf(m, BETA, s1);                 // mem1*BETA + syn1(old)
    syn1[row]  = fmaf(s1, ALPHA, cur);              // syn1*ALPHA + cur
    snext[row] = newout;
  }
}

// ---------------------------------------------------------------------------
// Phase 2b: readout layer, one time step. One wave per output neuron.
// Records mem2_new = mem2*BETA + syn2(old) BEFORE integrating ro_cur.
// ---------------------------------------------------------------------------
__global__ __launch_bounds__(NO * 32) void snn_readout(const float* __restrict__ Wro,
                                                       const float* __restrict__ s,
                                                       float* __restrict__ mem2,
                                                       float* __restrict__ syn2,
                                                       float* __restrict__ out,
                                                       int t) {
  const int o    = threadIdx.x >> 5;
  const int lane = threadIdx.x & 31;

  const float4* w4 = (const float4*)(Wro + (size_t)o * NH);
  const float4* s4 = (const float4*)s;

  float acc = 0.0f;
#pragma unroll 4
  for (int i = lane; i < NH / 4; i += 32) {
    const float4 w  = w4[i];
    const float4 sv = s4[i];
    acc = fmaf(w.x, sv.x, acc);
    acc = fmaf(w.y, sv.y, acc);
    acc = fmaf(w.z, sv.z, acc);
    acc = fmaf(w.w, sv.w, acc);
  }
#pragma unroll
  for (int off = 16; off > 0; off >>= 1) acc += __shfl_xor(acc, off, 32);

  if (lane == 0) {
    const float s2 = syn2[o];
    const float m2 = fmaf(mem2[o], BETA, s2);       // recorded value
    out[(size_t)o * T_STEPS + t] = m2;              // output is [NO, T]
    syn2[o] = fmaf(s2, ALPHA, acc);
    mem2[o] = m2;
  }
}

// ---------------------------------------------------------------------------
extern "C" void kernel_launch(void* const* d_in, const int* in_sizes, int n_in,
                              void* d_out, int out_size, void* d_ws, size_t ws_size,
                              hipStream_t stream) {
  (void)in_sizes; (void)n_in; (void)out_size; (void)ws_size;

  const float* spikes = (const float*)d_in[0];   // [T, NI]
  const float* Win    = (const float*)d_in[1];   // [NH, NI]
  const float* Wfb    = (const float*)d_in[2];   // [NH, NH]
  const float* Wro    = (const float*)d_in[3];   // [NO, NH]
  float*       out    = (float*)d_out;           // [NO, T]

  // Workspace layout (floats): Cin[NH*T] | mem1[NH] syn1[NH] out1a[NH] out1b[NH] mem2[NO] syn2[NO]
  float* Cin   = (float*)d_ws;
  float* mem1  = Cin   + (size_t)NH * T_STEPS;
  float* syn1  = mem1  + NH;
  float* out1a = syn1  + NH;
  float* out1b = out1a + NH;
  float* mem2  = out1b + NH;
  float* syn2  = mem2  + NO;

  // Phase 0: zero LIF state (contiguous block starting at mem1)
  const int stateN = 4 * NH + 2 * NO;
  snn_init<<<(stateN + 255) / 256, 256, 0, stream>>>(mem1, stateN);

  // Phase 1: precompute all input currents with fp32 WMMA (parallel over T)
  dim3 g1(NH / 16, (T_STEPS + 15) / 16);
  snn_in_gemm<<<g1, 32, 0, stream>>>(Win, spikes, Cin);

  // Phase 2: sequential recurrence; per-step grid sync via kernel boundaries
  for (int t = 0; t < T_STEPS; ++t) {
    const float* sp = (t & 1) ? out1b : out1a;
    float*       sn = (t & 1) ? out1a : out1b;
    snn_hidden<<<NH / 8, 256, 0, stream>>>(Wfb, Cin, mem1, syn1, sp, sn, t);
    snn_readout<<<1, NO * 32, 0, stream>>>(Wro, sn, mem2, syn2, out, t);
  }
}